// HungarianMatcher_71571335020565
// MI455X (gfx1250) — compile-verified
//
#include <hip/hip_runtime.h>

typedef __attribute__((ext_vector_type(2))) float v2f;
typedef __attribute__((ext_vector_type(8))) float v8f;

#define GAMMA_ 2.0f
#define EPS_ 1e-8f
#define COST_CLASS_ 2.0f
#define COST_GEOM_ 5.0f
#define WAVES_PER_BLOCK 8

// Kernel 1: per (row, class) focal cost table, pre-scaled by COST_CLASS.
// delta[n*2+c] = 2 * ( (1-p)^2 * -log(p+eps) - p^2 * -log(1-p+eps) ), p = sigmoid(logit)
__global__ __launch_bounds__(256) void delta_kernel(const float* __restrict__ logits,
                                                    float* __restrict__ delta,
                                                    int total) {
    int i = blockIdx.x * blockDim.x + threadIdx.x;
    if (i >= total) return;
    float x = logits[i];
    float p = 1.0f / (1.0f + expf(-x));
    float om = 1.0f - p;
    float pos = om * om * (-logf(p + EPS_));
    float neg = p * p * (-logf(om + EPS_));
    delta[i] = COST_CLASS_ * (pos - neg);
}

// Kernel 2: one wave32 computes one 16x16 tile of the cost matrix.
// Geometric L1 costs go into the f32 accumulator; the class-cost gather is a
// 16x2 @ 2x16 one-hot matmul folded in via v_wmma_f32_16x16x4_f32.
__global__ __launch_bounds__(WAVES_PER_BLOCK * 32) void cost_tile_kernel(
    const float* __restrict__ predS,   // [N,4] pred screws (hx,hy,tx,ty)
    const float* __restrict__ delta,   // [N,2] scaled focal cost table
    const int*   __restrict__ tgtL,    // [T] target labels
    const float* __restrict__ tgtS,    // [T,4] target screws
    float*       __restrict__ out,     // [N,T]
    int N, int T, int colTiles, int totalTiles)
{
    __shared__ float sPred[WAVES_PER_BLOCK][16][4];
    __shared__ float sDelta[WAVES_PER_BLOCK][16][2];
    __shared__ float sTgt[WAVES_PER_BLOCK][16][4];
    __shared__ int   sLbl[WAVES_PER_BLOCK][16];

    const int wave = threadIdx.x >> 5;
    const int lane = threadIdx.x & 31;
    const int tile = blockIdx.x * WAVES_PER_BLOCK + wave;
    const bool live = (tile < totalTiles);   // wave-uniform

    int rowBase = 0, colBase = 0;
    if (live) {
        const int rowTile = tile / colTiles;
        const int colTile = tile - rowTile * colTiles;
        rowBase = rowTile * 16;
        colBase = colTile * 16;
        if (lane < 16) {
            // low half-wave stages 16 prediction rows
            const int r = rowBase + lane;
            if (r < N) {
                sPred[wave][lane][0] = predS[4 * r + 0];
                sPred[wave][lane][1] = predS[4 * r + 1];
                sPred[wave][lane][2] = predS[4 * r + 2];
                sPred[wave][lane][3] = predS[4 * r + 3];
                sDelta[wave][lane][0] = delta[2 * r + 0];
                sDelta[wave][lane][1] = delta[2 * r + 1];
            } else {
                sPred[wave][lane][0] = 0.f; sPred[wave][lane][1] = 0.f;
                sPred[wave][lane][2] = 0.f; sPred[wave][lane][3] = 0.f;
                sDelta[wave][lane][0] = 0.f; sDelta[wave][lane][1] = 0.f;
            }
        } else {
            // high half-wave stages 16 targets
            const int t = lane - 16;
            const int c = colBase + t;
            if (c < T) {
                sTgt[wave][t][0] = tgtS[4 * c + 0];
                sTgt[wave][t][1] = tgtS[4 * c + 1];
                sTgt[wave][t][2] = tgtS[4 * c + 2];
                sTgt[wave][t][3] = tgtS[4 * c + 3];
                sLbl[wave][t]    = tgtL[c];
            } else {
                sTgt[wave][t][0] = 0.f; sTgt[wave][t][1] = 0.f;
                sTgt[wave][t][2] = 0.f; sTgt[wave][t][3] = 0.f;
                sLbl[wave][t]    = -1;
            }
        }
    }
    __syncthreads();
    if (!live) return;

    const int j  = lane & 15;   // tile column
    const int hi = lane >> 4;   // row half: M = v + 8*hi (C/D VGPR layout)

    const float thx = sTgt[wave][j][0], thy = sTgt[wave][j][1];
    const float ttx = sTgt[wave][j][2], tty = sTgt[wave][j][3];
    const float tmx = 0.5f * (thx + ttx), tmy = 0.5f * (thy + tty);

    v8f acc;
#pragma unroll
    for (int v = 0; v < 8; ++v) {
        const int m = v + 8 * hi;
        const float phx = sPred[wave][m][0], phy = sPred[wave][m][1];
        const float ptx = sPred[wave][m][2], pty = sPred[wave][m][3];
        const float pmx = 0.5f * (phx + ptx), pmy = 0.5f * (phy + pty);
        const float g = fabsf(pmx - tmx) + fabsf(pmy - tmy)   // midpoint L1
                      + fabsf(phx - thx) + fabsf(phy - thy)   // head L1
                      + fabsf(ptx - ttx) + fabsf(pty - tty);  // tip  L1
        acc[v] = COST_GEOM_ * g;
    }

    // A (16x4 f32): lanes 0-15 hold (K=0,K=1) of row M=lane; lanes 16-31 hold (K=2,K=3)=0
    v2f a;
    a.x = (lane < 16) ? sDelta[wave][lane][0] : 0.0f;
    a.y = (lane < 16) ? sDelta[wave][lane][1] : 0.0f;
    // B (4x16 one-hot): VGPR0 = rows K0/K2 across half-waves, VGPR1 = rows K1/K3
    const int lbl = sLbl[wave][j];
    v2f b;
    b.x = (lane < 16 && lbl == 0) ? 1.0f : 0.0f;  // K=0 row; K=2 row zero
    b.y = (lane < 16 && lbl == 1) ? 1.0f : 0.0f;  // K=1 row; K=3 row zero

    // D = A*B + C : folds 2*(pos-neg)[m, label[j]] into the geometric cost
    acc = __builtin_amdgcn_wmma_f32_16x16x4_f32(
        /*neg_a=*/false, a, /*neg_b=*/false, b,
        /*c_mod=*/(short)0, acc, /*reuse_a=*/false, /*reuse_b=*/false);

    // Single base-address setup shared by both store paths; rows advance by T.
    const int c = colBase + j;
    const int rowTop = rowBase + 8 * hi;
    float* p = out + (size_t)rowTop * (size_t)T + c;
    const bool fullTile = (rowBase + 16 <= N) && (colBase + 16 <= T);  // wave-uniform
    if (fullTile) {
        // Fast path: 8 unguarded coalesced stores.
#pragma unroll
        for (int v = 0; v < 8; ++v) {
            *p = acc[v];
            p += T;
        }
    } else {
        // Edge tiles: same pointer-increment addressing, cheap per-row predicate.
        const bool colOk = (c < T);
#pragma unroll
        for (int v = 0; v < 8; ++v) {
            if (colOk && (rowTop + v < N)) *p = acc[v];
            p += T;
        }
    }
}

extern "C" void kernel_launch(void* const* d_in, const int* in_sizes, int n_in,
                              void* d_out, int out_size, void* d_ws, size_t ws_size,
                              hipStream_t stream) {
    const float* pred_logits = (const float*)d_in[0];  // [bs,nq,nc]
    const float* pred_screws = (const float*)d_in[1];  // [bs,nq,4]
    const int*   tgt_labels  = (const int*)d_in[2];    // [T]
    const float* tgt_screws  = (const float*)d_in[3];  // [T,4]
    float*       out         = (float*)d_out;          // [N,T]

    const int N = in_sizes[1] / 4;       // bs*nq = 28800
    const int totalD = in_sizes[0];      // N*nc  = 57600
    const int T = in_sizes[2];           // 960

    float* delta = (float*)d_ws;         // N*nc floats of scratch

    delta_kernel<<<(totalD + 255) / 256, 256, 0, stream>>>(pred_logits, delta, totalD);

    const int rowTiles = (N + 15) / 16;
    const int colTiles = (T + 15) / 16;
    const int totalTiles = rowTiles * colTiles;
    const int blocks = (totalTiles + WAVES_PER_BLOCK - 1) / WAVES_PER_BLOCK;

    cost_tile_kernel<<<blocks, WAVES_PER_BLOCK * 32, 0, stream>>>(
        pred_screws, delta, tgt_labels, tgt_screws, out, N, T, colTiles, totalTiles);
}